// IzhikevichNet_6708738916573
// MI455X (gfx1250) — compile-verified
//
#include <hip/hip_runtime.h>
#include <hip/hip_bf16.h>
#include <math.h>

typedef __attribute__((ext_vector_type(2))) float v2f;
typedef __attribute__((ext_vector_type(8))) float v8f;

#define NWAVES 8
#define TILE_M 16
#define NHID 8192
#define NOUT 4096
#define NTOT 12288
#define NSTEPS 100

__device__ __forceinline__ float compress1(float y, float gain) {
    float s = (y > 0.f) ? 1.f : ((y < 0.f) ? -1.f : 0.f);
    return s * log1pf(gain * fabsf(y) + 1e-6f);
}

// y = W[M x K] @ x  (one 16-row tile per block, 8 waves split K, f32 WMMA accumulate)
// MODE 0: out[m] = compress(y[m], *gain_or_aux)
// MODE 1: out[m] = gain_or_aux[m] - y[m]        (feedback subtraction)
template <int MODE, int K>
__global__ __launch_bounds__(256) void gemv_wmma(const float* __restrict__ W,
                                                 const float* __restrict__ x,
                                                 const float* __restrict__ gain_or_aux,
                                                 float* __restrict__ out) {
    __shared__ float xs[K];                  // staged x
    __shared__ float part[NWAVES][TILE_M];   // per-wave partial row sums

    const int tid = threadIdx.x;

    // Cooperative, coalesced stage of x into LDS (K is a multiple of 1024).
    const float4* x4 = reinterpret_cast<const float4*>(x);
    float4* xs4 = reinterpret_cast<float4*>(xs);
    #pragma unroll
    for (int i = 0; i < K / 1024; ++i) xs4[tid + 256 * i] = x4[tid + 256 * i];
    __syncthreads();

    const int wave = tid >> 5;
    const int lane = tid & 31;
    const int r    = lane & 15;   // row within 16-row tile (A fragment: M = r)
    const int h    = lane >> 4;   // lane half selects k-pair {2h, 2h+1}
    const int row0 = blockIdx.x * TILE_M;
    const float* wrow = W + (long long)(row0 + r) * (long long)K + 2 * h;

    constexpr int KPW = K / NWAVES;          // K-slice per wave
    const int kbeg = wave * KPW;
    const int kend = kbeg + KPW;

    v8f acc0 = {0.f, 0.f, 0.f, 0.f, 0.f, 0.f, 0.f, 0.f};
    v8f acc1 = {0.f, 0.f, 0.f, 0.f, 0.f, 0.f, 0.f, 0.f};

    for (int k = kbeg; k < kend; k += 32) {  // 8 WMMA chunks (32 k) per iteration
        v2f a[8], b[8];
        #pragma unroll
        for (int j = 0; j < 8; ++j) {
            // A fragment: lane (r,h) holds W[row0+r][kc+2h], W[row0+r][kc+2h+1]
            a[j] = __builtin_nontemporal_load(
                reinterpret_cast<const v2f*>(wrow + k + 4 * j));  // streamed once -> NT
            // B fragment: x[kc + k_local] replicated across all 16 columns
            b[j] = *reinterpret_cast<const v2f*>(xs + k + 4 * j + 2 * h);
        }
        #pragma unroll
        for (int j = 0; j < 8; ++j) {
            if (j & 1)
                acc1 = __builtin_amdgcn_wmma_f32_16x16x4_f32(
                    false, a[j], false, b[j], (short)0, acc1, false, false);
            else
                acc0 = __builtin_amdgcn_wmma_f32_16x16x4_f32(
                    false, a[j], false, b[j], (short)0, acc0, false, false);
        }
    }
    v8f acc = acc0 + acc1;  // same C/D lane layout -> element-wise merge is valid

    // C/D layout: VGPR i, lanes 0-15 -> M=i ; lanes 16-31 -> M=8+i.
    // Every column N holds the same dot product; take column 0 (lanes 0 and 16).
    if (lane == 0) {
        #pragma unroll
        for (int i = 0; i < 8; ++i) part[wave][i] = acc[i];
    } else if (lane == 16) {
        #pragma unroll
        for (int i = 0; i < 8; ++i) part[wave][8 + i] = acc[i];
    }
    __syncthreads();

    if (tid < TILE_M) {
        float y = 0.f;
        #pragma unroll
        for (int w = 0; w < NWAVES; ++w) y += part[w][tid];
        if (MODE == 0) {
            out[row0 + tid] = compress1(y, gain_or_aux[0]);
        } else {
            out[row0 + tid] = gain_or_aux[row0 + tid] - y;
        }
    }
}

// Izhikevich dynamics: one thread per neuron, 100 steps, time-major coalesced stores.
__global__ __launch_bounds__(256) void izhi_kernel(const float* __restrict__ tot_h,
                                                   const float* __restrict__ iout,
                                                   float* __restrict__ vout,
                                                   float* __restrict__ uout) {
    int n = blockIdx.x * blockDim.x + threadIdx.x;
    if (n >= NTOT) return;
    const float I = (n < NHID) ? tot_h[n] : iout[n - NHID];
    float v = -65.f;
    float u = 0.2f * -65.f;
    __builtin_nontemporal_store(v, &vout[n]);
    __builtin_nontemporal_store(u, &uout[n]);
    for (int t = 1; t < NSTEPS; ++t) {
        float vn = v + (0.04f * v * v + 5.f * v + 0.14f - u + I);  // DT = 1
        float un = u + 0.02f * (0.2f * v - u);
        bool sp = vn >= 30.f;
        v = sp ? -65.f : vn;
        u = sp ? (un + 8.f) : un;
        __builtin_nontemporal_store(v, &vout[t * NTOT + n]);  // write-once trajectories
        __builtin_nontemporal_store(u, &uout[t * NTOT + n]);
    }
}

// loss = mean((iout - target)^2), single block, deterministic tree reduction.
__global__ __launch_bounds__(256) void loss_kernel(const float* __restrict__ iout,
                                                   const float* __restrict__ tgt,
                                                   float* __restrict__ out) {
    __shared__ float red[256];
    int tid = threadIdx.x;
    float s = 0.f;
    for (int i = tid; i < NOUT; i += 256) {
        float d = iout[i] - tgt[i];
        s += d * d;
    }
    red[tid] = s;
    __syncthreads();
    for (int off = 128; off > 0; off >>= 1) {
        if (tid < off) red[tid] += red[tid + off];
        __syncthreads();
    }
    if (tid == 0) out[0] = red[0] * (1.f / (float)NOUT);
}

extern "C" void kernel_launch(void* const* d_in, const int* in_sizes, int n_in,
                              void* d_out, int out_size, void* d_ws, size_t ws_size,
                              hipStream_t stream) {
    (void)in_sizes; (void)n_in; (void)out_size; (void)ws_size;

    const float* v_input = (const float*)d_in[0];
    const float* target  = (const float*)d_in[1];
    const float* W1      = (const float*)d_in[2];  // [8192 x 4096]
    const float* W2      = (const float*)d_in[3];  // [4096 x 8192]
    const float* Wfb     = (const float*)d_in[4];  // [8192 x 4096]
    const float* gain1   = (const float*)d_in[5];
    const float* gain2   = (const float*)d_in[6];

    float* ws    = (float*)d_ws;
    float* hid_c = ws;               // [8192] compressed hidden currents
    float* iout  = ws + NHID;        // [4096] compressed output currents
    float* tot_h = ws + NHID + NOUT; // [8192] hidden - feedback

    float* out  = (float*)d_out;
    float* vout = out + 1;
    float* uout = out + 1 + NSTEPS * NTOT;

    // 1) hid_c = compress(W1 @ v_input, gain1)      M=8192 K=4096
    gemv_wmma<0, 4096><<<NHID / TILE_M, 256, 0, stream>>>(W1, v_input, gain1, hid_c);
    // 2) iout  = compress(W2 @ hid_c, gain2)        M=4096 K=8192
    gemv_wmma<0, 8192><<<NOUT / TILE_M, 256, 0, stream>>>(W2, hid_c, gain2, iout);
    // 3) tot_h = hid_c - Wfb @ iout                 M=8192 K=4096
    gemv_wmma<1, 4096><<<NHID / TILE_M, 256, 0, stream>>>(Wfb, iout, hid_c, tot_h);
    // 4) Izhikevich dynamics -> v_vec, u_vec
    izhi_kernel<<<(NTOT + 255) / 256, 256, 0, stream>>>(tot_h, iout, vout, uout);
    // 5) loss
    loss_kernel<<<1, 256, 0, stream>>>(iout, target, out);
}